// OOSampler_30897994728000
// MI455X (gfx1250) — compile-verified
//
#include <hip/hip_runtime.h>
#include <stdint.h>

#define IMG     256
#define BATCH   64
#define CH      8
#define NROWS   (BATCH * IMG)          // 16384 (b,y) rows
#define LDS_STRIDE 260                 // dword stride per channel: avoids bank conflicts

// ---------------------------------------------------------------------------
// Kernel 1: per-(b,y) row, count masked pixels and find first masked x.
// Mask replicates reference exactly: f32 sqrt(dx*dx + dy*dy) < size/2 with
// separately-rounded mul/add (FP contraction disabled -> no FMA fusion).
// ---------------------------------------------------------------------------
__global__ void oos_count_rows(const int* __restrict__ sizes,
                               int* __restrict__ cnt,
                               int* __restrict__ x0a) {
#pragma clang fp contract(off)
    int id = blockIdx.x * blockDim.x + threadIdx.x;
    if (id >= NROWS) return;
    int b = id >> 8;
    int y = id & 255;
    float r   = 0.5f * (float)sizes[b];     // (s-1)/2 + 0.5 == s/2, exact in f32
    float dy  = (float)y - 127.5f;
    float dy2 = dy * dy;
    int c = 0, first = -1;
    for (int x = 0; x < IMG; ++x) {
        float dx = (float)x - 127.5f;
        float d2 = dx * dx + dy2;           // contract(off): mul then add, each RNE
        if (__fsqrt_rn(d2) < r) { if (first < 0) first = x; ++c; }
    }
    cnt[id] = c;
    x0a[id] = (first < 0) ? 0 : first;
}

// ---------------------------------------------------------------------------
// Kernel 2: exclusive scan of 16384 row counts in ONE 1024-thread workgroup.
// Each thread serially sums 16 counts, LDS Hillis-Steele over the 1024 partial
// sums, then writes its 16 exclusive offsets. offs[NROWS] = total N.
// ---------------------------------------------------------------------------
__global__ void oos_scan(const int* __restrict__ cnt, int* __restrict__ offs) {
    __shared__ int sh[1024];
    int t = threadIdx.x;
    int local[16];
    int sum = 0;
    #pragma unroll
    for (int j = 0; j < 16; ++j) { local[j] = cnt[t * 16 + j]; sum += local[j]; }
    sh[t] = sum;
    __syncthreads();
    for (int d = 1; d < 1024; d <<= 1) {
        int v = (t >= d) ? sh[t - d] : 0;
        __syncthreads();
        sh[t] += v;
        __syncthreads();
    }
    int run = sh[t] - sum;   // exclusive base for this thread's segment
    #pragma unroll
    for (int j = 0; j < 16; ++j) { offs[t * 16 + j] = run; run += local[j]; }
    if (t == 1023) offs[NROWS] = run;   // total N
}

// ---------------------------------------------------------------------------
// Kernel 3: one 256-thread block per (b,y) row. Stage the row's 8 channel
// segments into LDS with CDNA5 async global->LDS loads (ASYNCcnt), drain with
// s_wait_asynccnt, barrier, then each lane assembles its pixel's 8 channels
// from LDS (stride-260 dwords -> conflict-free over 64 banks) and emits two
// contiguous B128 stores (lane i writes 32B at base+32*i -> coalesced
// 1KB/wave), plus pts(y,x), imgid, batch_size.
// ---------------------------------------------------------------------------
__global__ void oos_emit(const float* __restrict__ batch,
                         const int* __restrict__ cnt,
                         const int* __restrict__ x0a,
                         const int* __restrict__ offs,
                         float* __restrict__ out) {
    __shared__ float smem[CH * LDS_STRIDE];
    int bid = blockIdx.x;
    int b = bid >> 8;
    int y = bid & 255;
    int n      = cnt[bid];
    int x0     = x0a[bid];
    int rowOff = offs[bid];
    int N      = offs[NROWS];
    int i = threadIdx.x;

    unsigned long long base = (unsigned long long)(uintptr_t)batch;

    if (i < n) {
        // byte offset of element (b, c=0, y, x0+i) in NCHW layout
        unsigned goff = ((unsigned)(b * CH) * (IMG * IMG)
                         + (unsigned)(y * IMG + x0 + i)) * 4u;
        #pragma unroll
        for (int c = 0; c < CH; ++c) {
            unsigned laddr = (unsigned)(uintptr_t)&smem[c * LDS_STRIDE + i];
            unsigned g = goff + (unsigned)c * (IMG * IMG * 4u);
            // VGLOBAL async load: VDST = LDS byte address, VADDR = i32 offset,
            // SADDR = 64-bit base. Tracked by ASYNCcnt.
            asm volatile("global_load_async_to_lds_b32 %0, %1, %2"
                         :: "v"(laddr), "v"(g), "s"(base)
                         : "memory");
        }
    }
    asm volatile("s_wait_asynccnt 0" ::: "memory");
    __syncthreads();

    if (i < n) {
        int row = rowOff + i;
        float4 t0 = make_float4(smem[0 * LDS_STRIDE + i], smem[1 * LDS_STRIDE + i],
                                smem[2 * LDS_STRIDE + i], smem[3 * LDS_STRIDE + i]);
        float4 t1 = make_float4(smem[4 * LDS_STRIDE + i], smem[5 * LDS_STRIDE + i],
                                smem[6 * LDS_STRIDE + i], smem[7 * LDS_STRIDE + i]);
        float4* texp = (float4*)(out + (size_t)row * 8);
        texp[0] = t0;
        texp[1] = t1;
        // pts: (y, x) order, float32
        float2* ptsp = (float2*)(out + (size_t)N * 8 + (size_t)row * 2);
        *ptsp = make_float2((float)y, (float)(x0 + i));
        // imgid
        out[(size_t)N * 10 + row] = (float)b;
    }
    if (bid == 0 && i == 0) out[(size_t)N * 11] = (float)BATCH;   // batch_size
}

// ---------------------------------------------------------------------------
extern "C" void kernel_launch(void* const* d_in, const int* in_sizes, int n_in,
                              void* d_out, int out_size, void* d_ws, size_t ws_size,
                              hipStream_t stream) {
    const float* batch = (const float*)d_in[0];   // [64, 8, 256, 256] f32
    const int*   sizes = (const int*)d_in[1];     // [64] i32
    float*       out   = (float*)d_out;

    int* cnt  = (int*)d_ws;            // NROWS
    int* x0a  = cnt + NROWS;           // NROWS
    int* offs = x0a + NROWS;           // NROWS + 1

    oos_count_rows<<<NROWS / 256, 256, 0, stream>>>(sizes, cnt, x0a);
    oos_scan<<<1, 1024, 0, stream>>>(cnt, offs);
    oos_emit<<<NROWS, 256, 0, stream>>>(batch, cnt, x0a, offs, out);
}